// HybridLSTM_65481071404385
// MI455X (gfx1250) — compile-verified
//
#include <hip/hip_runtime.h>
#include <hip/hip_bf16.h>

typedef float v2f __attribute__((ext_vector_type(2)));
typedef float v8f __attribute__((ext_vector_type(8)));

#define TDIM 128
#define BDIM 2048
#define DDIM 32
#define HDIM 4
#define DIN  36   // D + H

// ---------------------------------------------------------------------------
// Kernel 1: pre[r][n] = sum_{k<32} X[r][k] * Wall[n][k] + ball[n]
//   rows r = flattened (t, b), n = gate*4 + j  (gates: f,i,u,o)
//   One wave computes a 16x16 tile with 8x V_WMMA_F32_16X16X4_F32 (K=32).
// ---------------------------------------------------------------------------
__global__ __launch_bounds__(256) void qlstm_gemm(
    const float* __restrict__ X,
    const float* __restrict__ Wf, const float* __restrict__ bf,
    const float* __restrict__ Wi, const float* __restrict__ bi,
    const float* __restrict__ Wu, const float* __restrict__ bu,
    const float* __restrict__ Wo, const float* __restrict__ bo,
    float* __restrict__ pre, long long Mrows)
{
    long long wave  = ((long long)blockIdx.x * blockDim.x + threadIdx.x) >> 5;
    int       lane  = threadIdx.x & 31;
    long long rbase = wave * 16;
    if (rbase >= Mrows) return;            // whole-wave uniform: EXEC stays all-1

    const int n  = lane & 15;              // output column (gate*4 + j)
    const int hi = lane >> 4;              // half-wave selector

    const float* Wt[4] = {Wf, Wi, Wu, Wo};
    const float* bt[4] = {bf, bi, bu, bo};
    const float* wrow = Wt[n >> 2] + (n & 3) * DIN;   // row n of stacked W, len 36
    const float  bias = bt[n >> 2][n & 3];

    // C/D layout: VGPR v holds (M = v + 8*hi, N = lane&15); bias depends on N only
    v8f acc;
    #pragma unroll
    for (int v = 0; v < 8; ++v) acc[v] = bias;

    // A layout (16x4 f32): lane&15 = M row; VGPR0/1 = K = 2*hi, 2*hi+1
    const float* arow = X + (rbase + n) * DDIM;

    #pragma unroll
    for (int kc = 0; kc < 8; ++kc) {
        const int k = kc * 4 + hi * 2;                 // even -> 8B aligned
        v2f a = *(const v2f*)(arow + k);               // A: X[row][k..k+1]
        v2f b = *(const v2f*)(wrow + k);               // B: Wall[n][k..k+1]
        acc = __builtin_amdgcn_wmma_f32_16x16x4_f32(
            /*neg_a=*/false, a, /*neg_b=*/false, b,
            /*c_mod=*/(short)0, acc, /*reuse_a=*/false, /*reuse_b=*/false);
    }

    #pragma unroll
    for (int v = 0; v < 8; ++v) {
        long long row = rbase + v + hi * 8;
        pre[row * 16 + n] = acc[v];
    }
}

// ---------------------------------------------------------------------------
// Kernel 2: sequential LSTM recurrence. One thread per batch element.
//   qlayer(x)_j = prod_{k<=j} cos^2(x_k)   (closed form; RZ params cancel)
// ---------------------------------------------------------------------------
__device__ __forceinline__ float sigmoidf_(float x) {
    return 1.0f / (1.0f + __expf(-x));
}

__global__ __launch_bounds__(256) void qlstm_recur(
    const float* __restrict__ pre,
    const float* __restrict__ Wf, const float* __restrict__ Wi,
    const float* __restrict__ Wu, const float* __restrict__ Wo,
    float* __restrict__ out, float* __restrict__ state,
    int t0, int tc, int lastFlag)
{
    __shared__ float sWh[4][4][4];         // [gate][j][k]: W[gate][j][32+k]
    const int tid = threadIdx.x;
    if (tid < 64) {
        int g = tid >> 4, j = (tid >> 2) & 3, k = tid & 3;
        const float* W = (g == 0) ? Wf : (g == 1) ? Wi : (g == 2) ? Wu : Wo;
        sWh[g][j][k] = W[j * DIN + DDIM + k];
    }
    __syncthreads();

    const int b = blockIdx.x * blockDim.x + tid;
    if (b >= BDIM) return;

    float h[4], c[4];
    if (t0 == 0) {
        #pragma unroll
        for (int j = 0; j < 4; ++j) { h[j] = 0.0f; c[j] = 0.0f; }
    } else {
        float4 hv = *(const float4*)(state + (long long)b * 4);
        float4 cv = *(const float4*)(state + (long long)BDIM * 4 + (long long)b * 4);
        h[0] = hv.x; h[1] = hv.y; h[2] = hv.z; h[3] = hv.w;
        c[0] = cv.x; c[1] = cv.y; c[2] = cv.z; c[3] = cv.w;
    }

    for (int tt = 0; tt < tc; ++tt) {
        const float* p = pre + ((long long)tt * BDIM + b) * 16;
        float pv[16];
        #pragma unroll
        for (int i = 0; i < 16; ++i) pv[i] = p[i];     // 4x b128 coalesced

        float act[4][4];
        #pragma unroll
        for (int g = 0; g < 4; ++g) {
            float q = 1.0f;
            #pragma unroll
            for (int j = 0; j < 4; ++j) {
                float ang = pv[g * 4 + j]
                          + h[0] * sWh[g][j][0] + h[1] * sWh[g][j][1]
                          + h[2] * sWh[g][j][2] + h[3] * sWh[g][j][3];
                float z = __cosf(ang);
                q *= z * z;                             // prefix prod of cos^2
                act[g][j] = q;
            }
        }
        #pragma unroll
        for (int j = 0; j < 4; ++j) {
            float fg = sigmoidf_(act[0][j]);
            float ig = sigmoidf_(act[1][j]);
            float gg = tanhf(act[2][j]);
            float og = sigmoidf_(act[3][j]);
            c[j] = fg * c[j] + ig * gg;
            h[j] = og * tanhf(c[j]);
        }
        *(float4*)(out + ((long long)(t0 + tt) * BDIM + b) * 4) =
            make_float4(h[0], h[1], h[2], h[3]);
    }

    // persist carry for the next chunk
    *(float4*)(state + (long long)b * 4) = make_float4(h[0], h[1], h[2], h[3]);
    *(float4*)(state + (long long)BDIM * 4 + (long long)b * 4) =
        make_float4(c[0], c[1], c[2], c[3]);

    if (lastFlag) {
        float* hx = out + (long long)TDIM * BDIM * HDIM;
        float* cx = hx + (long long)BDIM * HDIM;
        *(float4*)(hx + (long long)b * 4) = make_float4(h[0], h[1], h[2], h[3]);
        *(float4*)(cx + (long long)b * 4) = make_float4(c[0], c[1], c[2], c[3]);
    }
}

// ---------------------------------------------------------------------------
extern "C" void kernel_launch(void* const* d_in, const int* in_sizes, int n_in,
                              void* d_out, int out_size, void* d_ws, size_t ws_size,
                              hipStream_t stream)
{
    (void)in_sizes; (void)n_in; (void)out_size;
    const float* X  = (const float*)d_in[0];
    const float* Wf = (const float*)d_in[1];
    const float* bf = (const float*)d_in[2];
    const float* Wi = (const float*)d_in[4];
    const float* bi = (const float*)d_in[5];
    const float* Wu = (const float*)d_in[7];
    const float* bu = (const float*)d_in[8];
    const float* Wo = (const float*)d_in[10];
    const float* bo = (const float*)d_in[11];
    float* out = (float*)d_out;

    // ws layout: [h,c carry state : 2*B*4 floats][pre chunk buffer]
    float*  state      = (float*)d_ws;
    size_t  stateBytes = (size_t)2 * BDIM * HDIM * sizeof(float);   // 64 KB
    float*  pre        = (float*)((char*)d_ws + stateBytes);
    size_t  avail      = (ws_size > stateBytes) ? (ws_size - stateBytes) : 0;
    size_t  perStep    = (size_t)BDIM * 16 * sizeof(float);         // 128 KB / t
    int TC = (int)(avail / perStep);
    if (TC < 1)    TC = 1;
    if (TC > TDIM) TC = TDIM;

    for (int t0 = 0; t0 < TDIM; t0 += TC) {
        const int tc = (TDIM - t0 < TC) ? (TDIM - t0) : TC;
        const long long Mrows = (long long)tc * BDIM;      // multiple of 16
        const int waves  = (int)(Mrows / 16);
        const int blocks = (waves + 7) / 8;                // 8 waves / 256-thr block
        qlstm_gemm<<<blocks, 256, 0, stream>>>(
            X + (long long)t0 * BDIM * DDIM,
            Wf, bf, Wi, bi, Wu, bu, Wo, bo, pre, Mrows);

        const int last = (t0 + tc >= TDIM) ? 1 : 0;
        qlstm_recur<<<(BDIM + 255) / 256, 256, 0, stream>>>(
            pre, Wf, Wi, Wu, Wo, out, state, t0, tc, last);
    }
}